// DilateAttention_32418413150925
// MI455X (gfx1250) — compile-verified
//
#include <hip/hip_runtime.h>

// DilateAttention fused kernel for MI455X (gfx1250).
// B=16, d=32, H=W=128, 3x3 window, dilation 2, zero 'same' padding.
//
// Memory-bound (AI ~2.3 flop/byte, 23.3 TB/s HBM -> ~6us floor):
// - K/V halo tiles (20x20x32 each) staged via gfx1250 async global->LDS
//   loads (ASYNCcnt / s_wait_asynccnt). 114.7KB LDS/WG, 2 WGs per 320KB WGP.
// - QK scores computed EXACTLY in fp32 on the matrix pipe with chained
//   V_WMMA_F32_16X16X4_F32: per pixel row, S[w,c] = sum_d q[d,w]*k[d,c] is a
//   true shared-operand matmul; the needed 9-tap band is extracted per lane.
// - Softmax + AV reduction fused, one global pass.

#define HEAD_DIM 32
#define IMG_H 128
#define IMG_W 128
#define BATCH 16
#define TS 16            // output tile side
#define HALO 2           // dilation * (kernel-1) / 2
#define TW (TS + 2*HALO) // 20: staged tile side with halo
#define NTAPS 9
#define ATTN_SCALE 0.17677669529663687f  // 32^-0.5

typedef __attribute__((ext_vector_type(2))) float v2f;
typedef __attribute__((ext_vector_type(8))) float v8f;

__device__ __forceinline__ unsigned lds_off32(const void* p) {
    // generic (flat) pointer to LDS: low 32 bits carry the LDS byte offset
    return (unsigned)(unsigned long long)p;
}

__device__ __forceinline__ void async_ld_f32(unsigned lds_byte_off, const float* gptr) {
    // GLOBAL_LOAD_ASYNC_TO_LDS_B32 (GV mode): VDST = LDS byte offset VGPR,
    // VADDR = 64-bit global address. Tracked by ASYNCcnt, honors EXEC.
    asm volatile("global_load_async_to_lds_b32 %0, %1, off"
                 :: "v"(lds_byte_off), "v"(gptr)
                 : "memory");
}

__device__ __forceinline__ void wait_async0() {
    asm volatile("s_wait_asynccnt 0" ::: "memory");
}

__device__ __forceinline__ void wait_ds0() {
    asm volatile("s_wait_dscnt 0" ::: "memory");
}

extern "C" __global__ __launch_bounds__(256)
void dilate_attn_fused(const float* __restrict__ q,
                       const float* __restrict__ k,
                       const float* __restrict__ v,
                       float* __restrict__ out)
{
    extern __shared__ float lds[];
    float* k_lds  = lds;                          // [32][20][20] = 12800 f
    float* v_lds  = lds + HEAD_DIM * TW * TW;     // [32][20][20] = 12800 f
    float* sc_lds = v_lds + HEAD_DIM * TW * TW;   // [16 py][3 i][16 w][4] = 3072 f

    const int b  = blockIdx.z;
    const int ty = blockIdx.y * TS;               // tile origin in image
    const int tx = blockIdx.x * TS;
    const int tid = threadIdx.x;                  // 0..255

    // ---------------- stage K and V halo tiles into LDS ----------------
    // In-bounds elements: async global->LDS (ASYNCcnt). OOB halo: zero via DS
    // stores (disjoint addresses, so no ASYNC/DS ordering hazard).
    const int total = HEAD_DIM * TW * TW;         // 12800 -> 50 per thread
    for (int e = tid; e < total; e += 256) {
        const int d  = e / (TW * TW);
        const int r  = e - d * (TW * TW);
        const int y  = r / TW;
        const int x  = r - y * TW;
        const int gy = ty + y - HALO;
        const int gx = tx + x - HALO;
        const bool inb = (gy >= 0) && (gy < IMG_H) && (gx >= 0) && (gx < IMG_W);
        if (inb) {
            const long gidx = (((long)b * HEAD_DIM + d) * IMG_H + gy) * IMG_W + gx;
            async_ld_f32(lds_off32(k_lds + e), k + gidx);
            async_ld_f32(lds_off32(v_lds + e), v + gidx);
        } else {
            k_lds[e] = 0.0f;
            v_lds[e] = 0.0f;
        }
    }
    wait_async0();          // my wave's async copies landed in LDS
    __syncthreads();        // everyone's did

    // ---------------- QK scores on the WMMA pipe (exact fp32) ----------------
    // Wave W owns pixel rows py = 2W, 2W+1 (exactly its own lanes' pixels).
    // For each (py, tap-row i): S[w, c] = sum_{d<32} q[d,w] * k_lds[d][py+2i][c]
    // via 8 chained V_WMMA_F32_16X16X4_F32 (K=4 each), two 16-wide column tiles.
    //
    // Fragment layouts (wave32, f32):
    //   A 16x4 : lane l, vgpr p -> A[M=l%16, K=2*(l/16)+p]
    //   B 4x16 : lane l, vgpr p -> B[K=2*(l/16)+p, N=l%16]
    //   D 16x16: lane l, vgpr r -> D[M=r+8*(l/16), N=l%16]
    const int lane   = tid & 31;
    const int ln     = lane & 15;
    const int lh     = lane >> 4;
    const int waveId = tid >> 5;

    for (int pyi = 0; pyi < 2; ++pyi) {
        const int py = 2 * waveId + pyi;

        // A fragments: q row py, all 32 channels (16-wide coalesced loads)
        const float* qrow =
            q + ((long)b * HEAD_DIM * IMG_H + (ty + py)) * IMG_W + tx + ln;
        float afrag[8][2];
        #pragma unroll
        for (int s = 0; s < 8; ++s) {
            #pragma unroll
            for (int p = 0; p < 2; ++p) {
                afrag[s][p] = qrow[(long)(4 * s + 2 * lh + p) * (IMG_H * IMG_W)];
            }
        }

        #pragma unroll
        for (int i = 0; i < 3; ++i) {
            const int krow = py + 2 * i;
            #pragma unroll
            for (int t = 0; t < 2; ++t) {       // column tiles c = 16t + ln
                v8f acc = {0.f, 0.f, 0.f, 0.f, 0.f, 0.f, 0.f, 0.f};
                #pragma unroll
                for (int s = 0; s < 8; ++s) {
                    const int d0 = 4 * s + 2 * lh;
                    v2f A; A.x = afrag[s][0]; A.y = afrag[s][1];
                    v2f Bf;
                    Bf.x = k_lds[(d0       * TW + krow) * TW + 16 * t + ln];
                    Bf.y = k_lds[((d0 + 1) * TW + krow) * TW + 16 * t + ln];
                    acc = __builtin_amdgcn_wmma_f32_16x16x4_f32(
                        false, A, false, Bf, (short)0, acc, false, false);
                }
                // extract the 9-tap band: pixel w needs c - w in {0,2,4}
                #pragma unroll
                for (int r = 0; r < 8; ++r) {
                    const int w    = r + 8 * lh;
                    const int c    = 16 * t + ln;
                    const int diff = c - w;
                    if (diff >= 0 && diff <= 4 && (diff & 1) == 0) {
                        sc_lds[((py * 3 + i) * 16 + w) * 4 + (diff >> 1)] = acc[r];
                    }
                }
            }
        }
    }
    wait_ds0();   // same-wave DS in-order; fence before score readback

    // ---------------- per-pixel softmax + AV reduction ----------------
    const int px = tid & 15;
    const int py = tid >> 4;      // == pixel rows this wave just produced
    const int gy = ty + py;
    const int gx = tx + px;

    float s9[NTAPS];
    #pragma unroll
    for (int i = 0; i < 3; ++i) {
        #pragma unroll
        for (int j = 0; j < 3; ++j) {
            s9[i * 3 + j] = sc_lds[((py * 3 + i) * 16 + px) * 4 + j];
        }
    }

    // softmax over the 9 taps (scaled). OOB taps have exact score 0 from the
    // zero-filled halo, matching the reference's zero-padding semantics.
    float m = s9[0];
    #pragma unroll
    for (int n = 1; n < NTAPS; ++n) m = fmaxf(m, s9[n]);
    float wgt[NTAPS];
    float wsum = 0.0f;
    #pragma unroll
    for (int n = 0; n < NTAPS; ++n) {
        wgt[n] = __expf((s9[n] - m) * ATTN_SCALE);
        wsum += wgt[n];
    }
    const float inv = 1.0f / wsum;
    #pragma unroll
    for (int n = 0; n < NTAPS; ++n) wgt[n] *= inv;

    // weighted V reduction, write 32 channels (lane-coalesced stores)
    const long pix = ((long)b * HEAD_DIM * IMG_H + gy) * IMG_W + gx;
    #pragma unroll
    for (int d = 0; d < HEAD_DIM; ++d) {
        const float* vb = v_lds + (d * TW + py) * TW + px;
        float acc = 0.0f;
        #pragma unroll
        for (int i = 0; i < 3; ++i) {
            #pragma unroll
            for (int j = 0; j < 3; ++j) {
                acc = fmaf(wgt[i * 3 + j], vb[(2 * i) * TW + 2 * j], acc);
            }
        }
        out[pix + (long)d * (IMG_H * IMG_W)] = acc;
    }
}

extern "C" void kernel_launch(void* const* d_in, const int* in_sizes, int n_in,
                              void* d_out, int out_size, void* d_ws, size_t ws_size,
                              hipStream_t stream) {
    (void)in_sizes; (void)n_in; (void)d_ws; (void)ws_size; (void)out_size;
    const float* q = (const float*)d_in[0];
    const float* k = (const float*)d_in[1];
    const float* v = (const float*)d_in[2];
    float* out = (float*)d_out;

    const size_t smem =
        (2u * HEAD_DIM * TW * TW + 16u * 3u * 16u * 4u) * sizeof(float); // 114,688 B
    // allow >64KB dynamic LDS (gfx1250 WGP has 320KB); idempotent, capture-safe
    (void)hipFuncSetAttribute((const void*)dilate_attn_fused,
                              hipFuncAttributeMaxDynamicSharedMemorySize,
                              (int)smem);

    dim3 grid(IMG_W / TS, IMG_H / TS, BATCH);   // 8 x 8 x 16
    dim3 block(256, 1, 1);                       // 8 wave32s
    dilate_attn_fused<<<grid, block, smem, stream>>>(q, k, v, out);
}